// ReconstructionOriginalLayer_28802050687291
// MI455X (gfx1250) — compile-verified
//
#include <hip/hip_runtime.h>
#include <math.h>

// ============================================================================
// Phase-comp Doppler pipeline for MI455X (gfx1250), fp32 WMMA path.
//
// Workspace layout (floats). Requires ws_size >= ~68.3 MB.
//   [0,      PLANE)  xr plane, layout [b][c][t]  (x2 -> ifft result -> o, in place)
//   [PLANE, 2PLANE)  xi plane
//   then: DFT64 (64x64 r/i), twiddle4096 (64x64 r/i), DFT512 (512x256 r/i)
// ============================================================================

typedef __attribute__((ext_vector_type(2))) float v2f;
typedef __attribute__((ext_vector_type(8))) float v8f;

static constexpr int    BN = 8;       // batches
static constexpr int    TN = 4096;    // FR0
static constexpr int    CN = 256;     // channels
static constexpr int    FN = 512;     // FR1
static constexpr size_t PLANE = (size_t)BN * CN * TN;

static constexpr size_t OFF_XR   = 0;
static constexpr size_t OFF_XI   = PLANE;
static constexpr size_t OFF_W64R = 2 * PLANE;
static constexpr size_t OFF_W64I = OFF_W64R + 64 * 64;
static constexpr size_t OFF_TWR  = OFF_W64I + 64 * 64;
static constexpr size_t OFF_TWI  = OFF_TWR + 64 * 64;
static constexpr size_t OFF_F512R = OFF_TWI + 64 * 64;
static constexpr size_t OFF_F512I = OFF_F512R + (size_t)FN * CN;

// XOR-ish swizzle for 64x64 LDS tiles: row-contiguous AND column accesses
// hit distinct banks (bank = (c+r)&63), no padding so 4 tiles = exactly 64KB.
__device__ __forceinline__ int sw(int r, int c) { return r * 64 + ((c + r) & 63); }

// V_WMMA_F32_16X16X4_F32: A 16x4 (2 VGPR), B 4x16 (2 VGPR), C/D 16x16 f32 (8 VGPR).
// 8-arg pattern per probe doc: (neg_a, A, neg_b, B, c_mod, C, reuse_a, reuse_b).
// f32 WMMA NEG applies only to C, so pass negated operand data for the -Ai*Bi term.
__device__ __forceinline__ v8f wmma4(v2f a, v2f b, v8f c) {
  return __builtin_amdgcn_wmma_f32_16x16x4_f32(false, a, false, b, (short)0, c,
                                               false, false);
}

// Complex FMA tile: (cr + i*ci) += (ar + i*ai) * (br + i*bi), 4 real WMMAs.
__device__ __forceinline__ void cwmma(v2f ar, v2f ai, v2f br, v2f bi,
                                      v8f& cr, v8f& ci) {
  cr = wmma4(ar, br, cr);
  v2f nai = -ai;
  cr = wmma4(nai, bi, cr);
  ci = wmma4(ar, bi, ci);
  ci = wmma4(ai, br, ci);
}

// A-frag (16x4, M x K): lane holds M = lane&15, K = 2*(lane>>4) + {0,1}.
// B-frag (4x16, K x N): lane holds N = lane&15, K = 2*(lane>>4) + {0,1}.
// C/D (16x16): vgpr r -> row = r + 8*(lane>>4), col = lane&15.

// ----------------------------------------------------------------------------
// Twiddle tables. Integer angle reduction mod N keeps cosf/sinf args in [0,2pi).
// ----------------------------------------------------------------------------
__global__ void k_init_tables(float* __restrict__ ws) {
  int idx = blockIdx.x * blockDim.x + threadIdx.x;
  const float twopi = 6.283185307179586f;
  if (idx < 64 * 64) {
    int i = idx >> 6, j = idx & 63;
    float th = twopi * (float)((i * j) & 63) * (1.0f / 64.0f);
    ws[OFF_W64R + idx] = cosf(th);        // inverse-FFT sign: e^{+i theta}
    ws[OFF_W64I + idx] = sinf(th);
    float th2 = twopi * (float)((i * j) & 4095) * (1.0f / 4096.0f);
    ws[OFF_TWR + idx] = cosf(th2);
    ws[OFF_TWI + idx] = sinf(th2);
  }
  if (idx < FN * CN) {
    int f = idx >> 8, n = idx & 255;
    float th = twopi * (float)((f * n) & 511) * (1.0f / 512.0f);
    ws[OFF_F512R + idx] = cosf(th);       // forward-FFT sign: e^{-i theta}
    ws[OFF_F512I + idx] = -sinf(th);
  }
}

// ----------------------------------------------------------------------------
// Pass-through x output + scale by 1/(MASK+RHO) + transpose (t,c)->(c,t) planes.
// ----------------------------------------------------------------------------
__global__ void k_pass_scale_transpose(const float* __restrict__ xr,
                                       const float* __restrict__ xi,
                                       float2* __restrict__ x_out,
                                       float* __restrict__ wr,
                                       float* __restrict__ wi) {
  __shared__ float tr[32][33];
  __shared__ float ti[32][33];
  int b = blockIdx.z;
  int t0 = blockIdx.x * 32;
  int c0 = blockIdx.y * 32;
  int tx = threadIdx.x, ty = threadIdx.y;          // 32 x 8
  const float s = 1.0f / 1.5f;                     // 1/(MASK+RHO)
  for (int i = 0; i < 4; ++i) {
    int t = t0 + ty + i * 8;
    int c = c0 + tx;
    size_t gi = ((size_t)b * TN + t) * CN + c;
    float re = xr[gi], im = xi[gi];
    x_out[gi] = make_float2(re, im);               // output #1: untouched x
    tr[ty + i * 8][tx] = re * s;
    ti[ty + i * 8][tx] = im * s;
  }
  __syncthreads();
  for (int i = 0; i < 4; ++i) {
    int c = c0 + ty + i * 8;
    int t = t0 + tx;
    size_t go = ((size_t)b * CN + c) * (size_t)TN + t;
    wr[go] = tr[tx][ty + i * 8];
    wi[go] = ti[tx][ty + i * 8];
  }
}

// ----------------------------------------------------------------------------
// 4096-pt IFFT per row as 4-step 64x64: X[k1][k2] -> T = X*W64 -> U = T.*tw
// -> O = U^T * W64, scaled 1/4096. Two complex 64^3 GEMMs on WMMA f32.
// One workgroup (8 waves) per row; 16 output tiles -> 2 tiles/wave.
// Dynamic LDS: Xr,Xi (reused for U) + Wr,Wi = 64KB.
// ----------------------------------------------------------------------------
__global__ void __launch_bounds__(256) k_ifft4096(
    float* __restrict__ wr, float* __restrict__ wi,
    const float* __restrict__ w64r, const float* __restrict__ w64i,
    const float* __restrict__ twr, const float* __restrict__ twi) {
  extern __shared__ float smem[];
  float* Xr = smem;               // 64x64: X, then U
  float* Xi = smem + 4096;
  float* Wr = smem + 8192;        // DFT64
  float* Wi = smem + 12288;
  size_t base = (size_t)blockIdx.x * TN;
  int tid = threadIdx.x, lane = tid & 31, wave = tid >> 5;

  for (int e = tid; e < 4096; e += 256) {
    int k1 = e & 63, k2 = e >> 6;                 // x[k1 + 64*k2]
    Xr[sw(k1, k2)] = wr[base + e];
    Xi[sw(k1, k2)] = wi[base + e];
    Wr[sw(e >> 6, e & 63)] = w64r[e];
    Wi[sw(e >> 6, e & 63)] = w64i[e];
  }
  __syncthreads();

  v8f uR[2], uI[2];
  // Stage 1: T[k1][m1] = sum_k2 X[k1][k2] * W64[k2][m1]; then twiddle.
  for (int rep = 0; rep < 2; ++rep) {
    int tId = wave * 2 + rep;
    int tm = (tId >> 2) * 16, tn = (tId & 3) * 16;
    v8f cr = {0, 0, 0, 0, 0, 0, 0, 0}, ci = {0, 0, 0, 0, 0, 0, 0, 0};
    int m = tm + (lane & 15), n = tn + (lane & 15);
    for (int kk = 0; kk < 64; kk += 4) {
      int kb = kk + 2 * (lane >> 4);
      v2f ar = {Xr[sw(m, kb)], Xr[sw(m, kb + 1)]};
      v2f ai = {Xi[sw(m, kb)], Xi[sw(m, kb + 1)]};
      v2f br = {Wr[sw(kb, n)], Wr[sw(kb + 1, n)]};
      v2f bi = {Wi[sw(kb, n)], Wi[sw(kb + 1, n)]};
      cwmma(ar, ai, br, bi, cr, ci);
    }
    int hi = lane >> 4;
    for (int r = 0; r < 8; ++r) {
      int k1 = tm + r + 8 * hi, m1 = tn + (lane & 15);
      float t_r = twr[k1 * 64 + m1], t_i = twi[k1 * 64 + m1];  // hot in L2
      uR[rep][r] = cr[r] * t_r - ci[r] * t_i;
      uI[rep][r] = cr[r] * t_i + ci[r] * t_r;
    }
  }
  __syncthreads();  // all stage-1 LDS reads done before overwriting X with U
  for (int rep = 0; rep < 2; ++rep) {
    int tId = wave * 2 + rep;
    int tm = (tId >> 2) * 16, tn = (tId & 3) * 16;
    int hi = lane >> 4;
    for (int r = 0; r < 8; ++r) {
      int k1 = tm + r + 8 * hi, m1 = tn + (lane & 15);
      Xr[sw(k1, m1)] = uR[rep][r];
      Xi[sw(k1, m1)] = uI[rep][r];
    }
  }
  __syncthreads();
  // Stage 2: O[m1][m2] = sum_k1 U[k1][m1] * W64[k1][m2]; X[m1+64*m2] = O/4096.
  const float inv_n = 1.0f / 4096.0f;
  for (int rep = 0; rep < 2; ++rep) {
    int tId = wave * 2 + rep;
    int tm = (tId >> 2) * 16, tn = (tId & 3) * 16;
    v8f cr = {0, 0, 0, 0, 0, 0, 0, 0}, ci = {0, 0, 0, 0, 0, 0, 0, 0};
    int m = tm + (lane & 15), n = tn + (lane & 15);
    for (int kk = 0; kk < 64; kk += 4) {
      int kb = kk + 2 * (lane >> 4);
      v2f ar = {Xr[sw(kb, m)], Xr[sw(kb + 1, m)]};   // A[m][k] = U[k][m]
      v2f ai = {Xi[sw(kb, m)], Xi[sw(kb + 1, m)]};
      v2f br = {Wr[sw(kb, n)], Wr[sw(kb + 1, n)]};
      v2f bi = {Wi[sw(kb, n)], Wi[sw(kb + 1, n)]};
      cwmma(ar, ai, br, bi, cr, ci);
    }
    int hi = lane >> 4;
    for (int r = 0; r < 8; ++r) {
      int m1 = tm + r + 8 * hi, m2 = tn + (lane & 15);
      size_t gi = base + (size_t)(m1 + 64 * m2);
      wr[gi] = cr[r] * inv_n;
      wi[gi] = ci[r] * inv_n;
    }
  }
}

// ----------------------------------------------------------------------------
// Sequential phase-comp scan: one persistent workgroup per batch (8 WGPs busy).
// pref lives in LDS (32 KB << 320 KB WGP LDS). o[n-8] re-read from global:
// it was written 8 iterations ago by this same block -> guaranteed L2 hit.
// In-place: plane row n is f[n] before step n and o[n] after.
// ----------------------------------------------------------------------------
__global__ void __launch_bounds__(1024) k_phase_recur(float* __restrict__ wr,
                                                      float* __restrict__ wi) {
  extern __shared__ float smem[];
  float* prefR = smem;            // 4096
  float* prefI = smem + 4096;     // 4096
  float* redR  = smem + 8192;     // 32 per-wave partials
  float* redI  = smem + 8224;     // 32
  float* ph    = smem + 8256;     // 2 (broadcast phase)
  int b = blockIdx.x;
  int tid = threadIdx.x, lane = tid & 31, wave = tid >> 5;
  float* pr_ = wr + (size_t)b * CN * TN;
  float* pi_ = wi + (size_t)b * CN * TN;

  for (int j = 0; j < 4; ++j) {              // pref = o_0 = final row 0
    int t = tid + j * 1024;
    prefR[t] = pr_[t];
    prefI[t] = pi_[t];
  }
  __syncthreads();

  for (int n = 1; n < CN; ++n) {
    size_t rb = (size_t)n * TN;
    float fr[4], fi[4], sR = 0.f, sI = 0.f;
    for (int j = 0; j < 4; ++j) {
      int t = tid + j * 1024;
      fr[j] = pr_[rb + t];
      fi[j] = pi_[rb + t];
      sR += prefR[t] * fr[j] + prefI[t] * fi[j];   // conj(pref)*f
      sI += prefR[t] * fi[j] - prefI[t] * fr[j];
    }
    if (n + 1 < CN) {                               // global_prefetch_b8
      __builtin_prefetch(&pr_[rb + TN + tid], 0, 0);
      __builtin_prefetch(&pi_[rb + TN + tid], 0, 0);
    }
    for (int off = 16; off > 0; off >>= 1) {        // wave32 reduce
      sR += __shfl_down(sR, off, 32);
      sI += __shfl_down(sI, off, 32);
    }
    if (lane == 0) { redR[wave] = sR; redI[wave] = sI; }
    __syncthreads();
    if (wave == 0) {
      float aR = redR[lane], aI = redI[lane];
      for (int off = 16; off > 0; off >>= 1) {
        aR += __shfl_down(aR, off, 32);
        aI += __shfl_down(aI, off, 32);
      }
      if (lane == 0) {
        float r = sqrtf(aR * aR + aI * aI);
        ph[0] = aR / r;                             // phase = s/|s|
        ph[1] = aI / r;
      }
    }
    __syncthreads();
    float cR = ph[0], cI = ph[1];
    for (int j = 0; j < 4; ++j) {
      int t = tid + j * 1024;
      float oR = fr[j] * cR + fi[j] * cI;           // o = f * conj(phase)
      float oI = fi[j] * cR - fr[j] * cI;
      float subR = 0.f, subI = 0.f;
      if (n >= 8) {                                  // window pop (0 for n<8)
        subR = pr_[(size_t)(n - 8) * TN + t];
        subI = pi_[(size_t)(n - 8) * TN + t];
      }
      prefR[t] += oR - subR;
      prefI[t] += oI - subI;
      pr_[rb + t] = oR;
      pi_[rb + t] = oI;
    }
    __threadfence();   // make o[n] visible for the read at step n+8
    __syncthreads();
  }
}

// ----------------------------------------------------------------------------
// Final zero-padded 512-pt DFT as complex GEMM: D[f][t] = sum_n F[f][n]*o[n][t],
// M=512 K=256 N=4096 per batch. Stores directly into the (b,t,f) output layout.
// Dynamic LDS: F chunk + O chunk = 64 KB, 4 K-chunks of 64.
// ----------------------------------------------------------------------------
__global__ void __launch_bounds__(256) k_fft512(
    const float* __restrict__ wr, const float* __restrict__ wi,
    const float* __restrict__ f512r, const float* __restrict__ f512i,
    float2* __restrict__ out3) {
  extern __shared__ float smem[];
  float* Fr = smem;
  float* Fi = smem + 4096;
  float* Or = smem + 8192;
  float* Oi = smem + 12288;
  int bid = blockIdx.x;                 // 8 * 8 * 64 blocks
  int b = bid >> 9;
  int rem = bid & 511;
  int fb = (rem >> 6) * 64;
  int tb = (rem & 63) * 64;
  int tid = threadIdx.x, lane = tid & 31, wave = tid >> 5;
  const float* plr = wr + (size_t)b * CN * TN;
  const float* pli = wi + (size_t)b * CN * TN;

  v8f zero = {0, 0, 0, 0, 0, 0, 0, 0};
  v8f cr[2] = {zero, zero}, ci[2] = {zero, zero};

  for (int n0 = 0; n0 < CN; n0 += 64) {
    __syncthreads();
    for (int e = tid; e < 4096; e += 256) {
      int i = e >> 6, j = e & 63;
      Fr[sw(i, j)] = f512r[(size_t)(fb + i) * CN + n0 + j];
      Fi[sw(i, j)] = f512i[(size_t)(fb + i) * CN + n0 + j];
      Or[sw(i, j)] = plr[(size_t)(n0 + i) * TN + tb + j];
      Oi[sw(i, j)] = pli[(size_t)(n0 + i) * TN + tb + j];
    }
    __syncthreads();
    for (int rep = 0; rep < 2; ++rep) {
      int tId = wave * 2 + rep;
      int tm = (tId >> 2) * 16, tn = (tId & 3) * 16;
      int m = tm + (lane & 15), nn = tn + (lane & 15);
      for (int kk = 0; kk < 64; kk += 4) {
        int kb = kk + 2 * (lane >> 4);
        v2f ar = {Fr[sw(m, kb)], Fr[sw(m, kb + 1)]};
        v2f ai = {Fi[sw(m, kb)], Fi[sw(m, kb + 1)]};
        v2f br = {Or[sw(kb, nn)], Or[sw(kb + 1, nn)]};
        v2f bi = {Oi[sw(kb, nn)], Oi[sw(kb + 1, nn)]};
        cwmma(ar, ai, br, bi, cr[rep], ci[rep]);
      }
    }
  }
  int hi = lane >> 4;
  for (int rep = 0; rep < 2; ++rep) {
    int tId = wave * 2 + rep;
    int tm = (tId >> 2) * 16, tn = (tId & 3) * 16;
    for (int r = 0; r < 8; ++r) {
      int f = fb + tm + r + 8 * hi;
      int t = tb + tn + (lane & 15);
      out3[((size_t)b * TN + t) * FN + f] = make_float2(cr[rep][r], ci[rep][r]);
    }
  }
}

// ----------------------------------------------------------------------------
extern "C" void kernel_launch(void* const* d_in, const int* in_sizes, int n_in,
                              void* d_out, int out_size, void* d_ws,
                              size_t ws_size, hipStream_t stream) {
  (void)in_sizes; (void)n_in; (void)out_size; (void)ws_size;
  const float* xr = (const float*)d_in[0];
  const float* xi = (const float*)d_in[1];
  float* ws = (float*)d_ws;
  float2* x_out = (float2*)d_out;                               // (8,4096,256) c64
  float2* out3 = (float2*)d_out + (size_t)BN * TN * CN;         // (8,4096,512) c64

  float* wsr = ws + OFF_XR;
  float* wsi = ws + OFF_XI;

  k_init_tables<<<(FN * CN + 255) / 256, 256, 0, stream>>>(ws);

  dim3 g0(TN / 32, CN / 32, BN), b0(32, 8);
  k_pass_scale_transpose<<<g0, b0, 0, stream>>>(xr, xi, x_out, wsr, wsi);

  k_ifft4096<<<BN * CN, 256, 4 * 4096 * sizeof(float), stream>>>(
      wsr, wsi, ws + OFF_W64R, ws + OFF_W64I, ws + OFF_TWR, ws + OFF_TWI);

  k_phase_recur<<<BN, 1024, (2 * 4096 + 64 + 2) * sizeof(float), stream>>>(wsr,
                                                                           wsi);

  k_fft512<<<BN * (FN / 64) * (TN / 64), 256, 4 * 4096 * sizeof(float),
             stream>>>(wsr, wsi, ws + OFF_F512R, ws + OFF_F512I, out3);
}